// QNNsubModel_9122510537272
// MI455X (gfx1250) — compile-verified
//
#include <hip/hip_runtime.h>
#include <hip/hip_bf16.h>

// ---------------------------------------------------------------------------
// QNN forward, reformulated:
//   psi = U_C * D * E2(x) * U_B * D * E1(x) * (U_A|0>)
// U_A|0> (256-vec), U_B, U_C (256x256 complex) depend only on phi -> built once
// per launch on device. Batch work = two fp32 real GEMMs [16384x512]x[512x512]
// on V_WMMA_F32_16X16X4_F32 (complex algebra expanded into the B matrix),
// plus cheap per-batch wave32 shuffle kernels for encode / measure.
// ---------------------------------------------------------------------------

typedef float v2f __attribute__((ext_vector_type(2)));
typedef float v8f __attribute__((ext_vector_type(8)));

// ----------------------------- complex helpers -----------------------------
__device__ __forceinline__ float2 cmul(float2 a, float2 b) {
  return make_float2(a.x * b.x - a.y * b.y, a.x * b.y + a.y * b.x);
}
__device__ __forceinline__ float2 cadd(float2 a, float2 b) {
  return make_float2(a.x + b.x, a.y + b.y);
}

struct m2 { float2 a00, a01, a10, a11; };

__device__ __forceinline__ m2 mmul(const m2& P, const m2& Q) {
  m2 R;
  R.a00 = cadd(cmul(P.a00, Q.a00), cmul(P.a01, Q.a10));
  R.a01 = cadd(cmul(P.a00, Q.a01), cmul(P.a01, Q.a11));
  R.a10 = cadd(cmul(P.a10, Q.a00), cmul(P.a11, Q.a10));
  R.a11 = cadd(cmul(P.a10, Q.a01), cmul(P.a11, Q.a11));
  return R;
}
__device__ __forceinline__ m2 rxm(float th) {
  float s, c; __sincosf(0.5f * th, &s, &c);
  return m2{make_float2(c, 0.f), make_float2(0.f, -s),
            make_float2(0.f, -s), make_float2(c, 0.f)};
}
__device__ __forceinline__ m2 rym(float th) {
  float s, c; __sincosf(0.5f * th, &s, &c);
  return m2{make_float2(c, 0.f), make_float2(-s, 0.f),
            make_float2(s, 0.f), make_float2(c, 0.f)};
}

// CZ-ring diagonal sign for flat index n (qubit i <-> bit 7-i)
__device__ __forceinline__ float cz_sign(int n) {
  int par = __popc((n & (n >> 1)) & 0x7f) + ((n & (n >> 7)) & 1);
  return (par & 1) ? -1.0f : 1.0f;
}

// --------------------- single-qubit gate on wave32 state --------------------
// State: 256 complex amps; amp index n = lane*8 + l  (lane 0..31, l 0..7).
// Qubit i acts on bit b = 7-i of n:  b<3 -> in-lane pairs, b>=3 -> shfl_xor.

__device__ __forceinline__ void apply_rx(float2 a[8], int b, float c, float s, int lane) {
  if (b < 3) {
    int m = 1 << b;
#pragma unroll
    for (int l = 0; l < 8; ++l) {
      if (l & m) continue;
      int h = l | m;
      float2 lo = a[l], hi = a[h];
      a[l] = make_float2(c * lo.x + s * hi.y, c * lo.y - s * hi.x);
      a[h] = make_float2(s * lo.y + c * hi.x, -s * lo.x + c * hi.y);
    }
  } else {
    int lm = 1 << (b - 3);
#pragma unroll
    for (int l = 0; l < 8; ++l) {
      float2 mine = a[l];
      float ox = __shfl_xor(mine.x, lm, 32);
      float oy = __shfl_xor(mine.y, lm, 32);
      a[l] = make_float2(c * mine.x + s * oy, c * mine.y - s * ox);  // RX symmetric
    }
  }
}

__device__ __forceinline__ void apply_ry(float2 a[8], int b, float c, float s, int lane) {
  if (b < 3) {
    int m = 1 << b;
#pragma unroll
    for (int l = 0; l < 8; ++l) {
      if (l & m) continue;
      int h = l | m;
      float2 lo = a[l], hi = a[h];
      a[l] = make_float2(c * lo.x - s * hi.x, c * lo.y - s * hi.y);
      a[h] = make_float2(s * lo.x + c * hi.x, s * lo.y + c * hi.y);
    }
  } else {
    int lm = 1 << (b - 3);
    float sg = (lane & lm) ? s : -s;
#pragma unroll
    for (int l = 0; l < 8; ++l) {
      float2 mine = a[l];
      float ox = __shfl_xor(mine.x, lm, 32);
      float oy = __shfl_xor(mine.y, lm, 32);
      a[l] = make_float2(c * mine.x + sg * ox, c * mine.y + sg * oy);
    }
  }
}

__device__ __forceinline__ void apply_u2(float2 a[8], int b, const m2& U, int lane) {
  if (b < 3) {
    int m = 1 << b;
#pragma unroll
    for (int l = 0; l < 8; ++l) {
      if (l & m) continue;
      int h = l | m;
      float2 lo = a[l], hi = a[h];
      a[l] = cadd(cmul(U.a00, lo), cmul(U.a01, hi));
      a[h] = cadd(cmul(U.a10, lo), cmul(U.a11, hi));
    }
  } else {
    int lm = 1 << (b - 3);
    bool hiS = (lane & lm) != 0;
    float2 ua = hiS ? U.a11 : U.a00;
    float2 ub = hiS ? U.a10 : U.a01;
#pragma unroll
    for (int l = 0; l < 8; ++l) {
      float2 mine = a[l];
      float2 oth = make_float2(__shfl_xor(mine.x, lm, 32), __shfl_xor(mine.y, lm, 32));
      a[l] = cadd(cmul(ua, mine), cmul(ub, oth));
    }
  }
}

// One variational layer: RX on all qubits, RY on all qubits, optional CZ ring.
__device__ __forceinline__ void qnn_layer(float2 a[8], const float* __restrict__ phi,
                                          int idx, const float sgn[8], int lane, bool cz) {
#pragma unroll
  for (int i = 0; i < 8; ++i) {
    float s, c; __sincosf(0.5f * phi[idx + i], &s, &c);
    apply_rx(a, 7 - i, c, s, lane);
  }
#pragma unroll
  for (int i = 0; i < 8; ++i) {
    float s, c; __sincosf(0.5f * phi[idx + 8 + i], &s, &c);
    apply_ry(a, 7 - i, c, s, lane);
  }
  if (cz) {
#pragma unroll
    for (int l = 0; l < 8; ++l) { a[l].x *= sgn[l]; a[l].y *= sgn[l]; }
  }
}

// Write column n0 of a 256x256 complex unitary into the packed real B-matrix:
// Wpack[q][j][kk], q=k>>2, j=0..511, kk=k&3, where the real 512x512 W encodes
// out_r = Ur*ar - Ui*ai ; out_i = Ui*ar + Ur*ai  (a interleaved re/im).
__device__ __forceinline__ void write_u(float* __restrict__ W, const float2 a[8],
                                        int n0, int lane) {
  float2* Wp = (float2*)W;
  size_t q = (size_t)(n0 >> 1);
  int half = n0 & 1;  // selects kk pair {0,1} or {2,3}
#pragma unroll
  for (int l = 0; l < 8; ++l) {
    int m = lane * 8 + l;
    float ur = a[l].x, ui = a[l].y;
    Wp[(q * 512 + 2 * m) * 2 + half]     = make_float2(ur, -ui);  // column j=2m
    Wp[(q * 512 + 2 * m + 1) * 2 + half] = make_float2(ui,  ur);  // column j=2m+1
  }
}

// ------------------------- kernel 1: build unitaries ------------------------
// block 0: v0 = (p1 layers)|0> ; blocks 1..256: U_B cols ; 257..512: U_C cols
__global__ __launch_bounds__(32) void qnn_build(const float* __restrict__ phi,
                                                float* __restrict__ v0,
                                                float* __restrict__ W2,
                                                float* __restrict__ W3) {
  int lane = threadIdx.x;
  int col = blockIdx.x;
  float2 a[8];
  float sgn[8];
#pragma unroll
  for (int l = 0; l < 8; ++l) sgn[l] = cz_sign(lane * 8 + l);

  int n0 = (col == 0) ? 0 : (col <= 256 ? col - 1 : col - 257);
#pragma unroll
  for (int l = 0; l < 8; ++l)
    a[l] = make_float2((lane * 8 + l) == n0 ? 1.0f : 0.0f, 0.0f);

  if (col == 0) {
    for (int t = 0; t < 200; ++t) qnn_layer(a, phi, 16 * t, sgn, lane, true);
    float2* o = (float2*)v0;
#pragma unroll
    for (int l = 0; l < 8; ++l) o[lane * 8 + l] = a[l];
  } else if (col <= 256) {
    for (int t = 0; t < 200; ++t) qnn_layer(a, phi, 3264 + 16 * t, sgn, lane, true);
    write_u(W2, a, n0, lane);
  } else {
    for (int t = 0; t < 199; ++t) qnn_layer(a, phi, 6528 + 16 * t, sgn, lane, true);
    qnn_layer(a, phi, 6528 + 16 * 199, sgn, lane, false);  // final layer, no CZ
    write_u(W3, a, n0, lane);
  }
}

// ---------------------- kernel 2/4: data encoding E(x) ----------------------
// One wave per batch element. 12 gates/qubit fused into a single 2x2 product,
// applied via shuffles; trailing CZ diagonal folded into the output A matrix.
__global__ __launch_bounds__(256) void qnn_encode(const float* __restrict__ src,
                                                  int perBatch,
                                                  const float* __restrict__ x,
                                                  const float* __restrict__ phi,
                                                  int base, int xoff,
                                                  float* __restrict__ Aout) {
  int lane = threadIdx.x & 31;
  int b = blockIdx.x * 8 + (threadIdx.x >> 5);
  const float2* s = (const float2*)(perBatch ? (src + (size_t)b * 512) : src);
  float2 a[8];
#pragma unroll
  for (int l = 0; l < 8; ++l) a[l] = s[lane * 8 + l];

#pragma unroll
  for (int i = 0; i < 8; ++i) {
    m2 M = {make_float2(1.f, 0.f), make_float2(0.f, 0.f),
            make_float2(0.f, 0.f), make_float2(1.f, 0.f)};
#pragma unroll
    for (int j = 0; j < 4; ++j) {
      float p0 = phi[base + 8 * i + 2 * j];
      float p1 = phi[base + 8 * i + 2 * j + 1];
      float xv = x[(size_t)b * 64 + xoff + 4 * i + j];
      m2 G = ((j & 1) == 0) ? mmul(rym(p1), mmul(rxm(xv), rym(p0)))
                            : mmul(rxm(p1), mmul(rym(xv), rxm(p0)));
      M = mmul(G, M);
    }
    apply_u2(a, 7 - i, M, lane);
  }

  float2* o = (float2*)(Aout + (size_t)b * 512);
#pragma unroll
  for (int l = 0; l < 8; ++l) {
    float sg = cz_sign(lane * 8 + l);
    o[lane * 8 + l] = make_float2(sg * a[l].x, sg * a[l].y);
  }
}

// ----------------------- kernel 3/5: fp32 WMMA GEMM -------------------------
// C[16384x512] = A[16384x512] * W[512x512], W packed as [q][j][kk] (k=4q+kk).
// Block = 4 waves, tile 32(M) x 512(N); A staged in 64KB LDS, quad-packed
// layout lds[q][m][kk] -> conflict-free 16x4 A-fragment reads. Each wave:
// 2 M-subtiles x 8 N-subtiles accumulators, 16 v_wmma_f32_16x16x4_f32 / K-quad.
__global__ __launch_bounds__(128) void qnn_gemm(const float* __restrict__ A,
                                                const float* __restrict__ W,
                                                float* __restrict__ C) {
  __shared__ float ldsA[32 * 512];  // 64 KB
  const int tid = threadIdx.x;
  const int wave = tid >> 5;
  const int lane = tid & 31;
  const int m0 = blockIdx.x * 32;

  {  // cooperative fill: lds[q][m][kk], float idx = q*128 + m*4 + kk
    const float4* Ag = (const float4*)(A + (size_t)m0 * 512);
    float4* Ls = (float4*)ldsA;
    const int m = tid >> 2;
    const int kp = (tid & 3) * 32;
#pragma unroll
    for (int i = 0; i < 32; ++i) {
      int kq = kp + i;
      Ls[kq * 32 + m] = Ag[m * 128 + kq];
    }
  }
  __syncthreads();

  const int n0 = wave * 128;
  const int l15 = lane & 15;
  const int kh = lane >> 4;  // K-half: lanes 0-15 -> kk 0..1, lanes 16-31 -> kk 2..3

  v8f zero = {0.f, 0.f, 0.f, 0.f, 0.f, 0.f, 0.f, 0.f};
  v8f acc[2][8];
#pragma unroll
  for (int ms = 0; ms < 2; ++ms)
#pragma unroll
    for (int ns = 0; ns < 8; ++ns) acc[ms][ns] = zero;

  const float2* Ls2 = (const float2*)ldsA;
  const float2* Wp = (const float2*)W;
  for (int q = 0; q < 128; ++q) {
    float2 t0 = Ls2[q * 64 + l15 * 2 + kh];          // A rows 0..15
    float2 t1 = Ls2[q * 64 + (l15 + 16) * 2 + kh];   // A rows 16..31
    v2f a0; a0.x = t0.x; a0.y = t0.y;
    v2f a1; a1.x = t1.x; a1.y = t1.y;
#pragma unroll
    for (int ns = 0; ns < 8; ++ns) {
      int j = n0 + ns * 16 + l15;
      float2 tb = Wp[((size_t)q * 512 + j) * 2 + kh];
      v2f bf; bf.x = tb.x; bf.y = tb.y;
      acc[0][ns] = __builtin_amdgcn_wmma_f32_16x16x4_f32(
          false, a0, false, bf, (short)0, acc[0][ns], false, false);
      acc[1][ns] = __builtin_amdgcn_wmma_f32_16x16x4_f32(
          false, a1, false, bf, (short)0, acc[1][ns], false, false);
    }
  }

#pragma unroll
  for (int ms = 0; ms < 2; ++ms)
#pragma unroll
    for (int ns = 0; ns < 8; ++ns) {
      int j = n0 + ns * 16 + l15;
#pragma unroll
      for (int v = 0; v < 8; ++v) {
        int m = m0 + ms * 16 + kh * 8 + v;  // ISA C/D layout: VGPR v -> M=v / v+8
        C[(size_t)m * 512 + j] = acc[ms][ns][v];
      }
    }
}

// --------------------------- kernel 6: measurement --------------------------
// <X_i> = sum_n Re(conj(psi_n) psi_{n^bit}) (bit 7-i -> lane xor 1<<(4-i));
// <Z_i> = sum_n (-1)^{bit} |psi_n|^2. Wave butterfly reduction, lane 0 stores.
__global__ __launch_bounds__(256) void qnn_measure(const float* __restrict__ S,
                                                   float* __restrict__ out) {
  int lane = threadIdx.x & 31;
  int b = blockIdx.x * 8 + (threadIdx.x >> 5);
  const float2* s = (const float2*)(S + (size_t)b * 512);
  float2 a[8];
#pragma unroll
  for (int l = 0; l < 8; ++l) a[l] = s[lane * 8 + l];

  float nrm = 0.f;
#pragma unroll
  for (int l = 0; l < 8; ++l) nrm += a[l].x * a[l].x + a[l].y * a[l].y;

  float res[8];
#pragma unroll
  for (int i = 0; i < 4; ++i) {
    int lm = 1 << (4 - i);
    float d = 0.f;
#pragma unroll
    for (int l = 0; l < 8; ++l) {
      float ox = __shfl_xor(a[l].x, lm, 32);
      float oy = __shfl_xor(a[l].y, lm, 32);
      d += a[l].x * ox + a[l].y * oy;
    }
    res[i] = d;                                        // X_i
    res[4 + i] = ((lane >> (4 - i)) & 1) ? -nrm : nrm; // Z_i
  }
#pragma unroll
  for (int k = 0; k < 8; ++k)
#pragma unroll
    for (int off = 16; off >= 1; off >>= 1)
      res[k] += __shfl_xor(res[k], off, 32);

  if (lane == 0) {
#pragma unroll
    for (int k = 0; k < 8; ++k) out[(size_t)b * 8 + k] = res[k];
  }
}

// ------------------------------- launcher -----------------------------------
extern "C" void kernel_launch(void* const* d_in, const int* in_sizes, int n_in,
                              void* d_out, int out_size, void* d_ws, size_t ws_size,
                              hipStream_t stream) {
  (void)in_sizes; (void)n_in; (void)out_size; (void)ws_size;
  const float* x = (const float*)d_in[0];    // [16384, 64]
  const float* phi = (const float*)d_in[1];  // [9728]
  float* ws = (float*)d_ws;

  // Workspace layout (floats): v0(1024 pad) | W2(256K) | W3(256K) | A(8M) | S(8M)
  float* v0 = ws;
  float* W2 = ws + 1024;
  float* W3 = W2 + 262144;
  float* bufA = W3 + 262144;
  float* bufS = bufA + (size_t)16384 * 512;  // total ~66 MB of d_ws

  qnn_build<<<513, 32, 0, stream>>>(phi, v0, W2, W3);
  qnn_encode<<<2048, 256, 0, stream>>>(v0, 0, x, phi, 3200, 0, bufA);
  qnn_gemm<<<512, 128, 0, stream>>>(bufA, W2, bufS);
  qnn_encode<<<2048, 256, 0, stream>>>(bufS, 1, x, phi, 6464, 32, bufA);
  qnn_gemm<<<512, 128, 0, stream>>>(bufA, W3, bufS);
  qnn_measure<<<2048, 256, 0, stream>>>(bufS, (float*)d_out);
}